// GeometricLoss_68573447847951
// MI455X (gfx1250) — compile-verified
//
#include <hip/hip_runtime.h>
#include <hip/hip_bf16.h>
#include <math.h>

typedef float v2f __attribute__((ext_vector_type(2)));
typedef float v8f __attribute__((ext_vector_type(8)));

#define NB 4
#define KNN 10
#define NPAIRS 6
#define SMAX 8

// ---------------------------------------------------------------------------
// Fundamental matrices for the 6 view pairs (one thread each).
// ---------------------------------------------------------------------------
__device__ static void cam_center(const float* __restrict__ P, float C[3]) {
    // C = -inv(P[:3,:3]) @ P[:3,3]   (P row-major 4x4)
    float m00 = P[0], m01 = P[1], m02 = P[2];
    float m10 = P[4], m11 = P[5], m12 = P[6];
    float m20 = P[8], m21 = P[9], m22 = P[10];
    float b0 = P[3], b1 = P[7], b2 = P[11];
    float c00 =  (m11 * m22 - m12 * m21);
    float c01 = -(m10 * m22 - m12 * m20);
    float c02 =  (m10 * m21 - m11 * m20);
    float det = m00 * c00 + m01 * c01 + m02 * c02;
    float id = 1.0f / det;
    float i00 = c00 * id, i01 = -(m01 * m22 - m02 * m21) * id, i02 = (m01 * m12 - m02 * m11) * id;
    float i10 = c01 * id, i11 =  (m00 * m22 - m02 * m20) * id, i12 = -(m00 * m12 - m02 * m10) * id;
    float i20 = c02 * id, i21 = -(m00 * m21 - m01 * m20) * id, i22 = (m00 * m11 - m01 * m10) * id;
    C[0] = -(i00 * b0 + i01 * b1 + i02 * b2);
    C[1] = -(i10 * b0 + i11 * b1 + i12 * b2);
    C[2] = -(i20 * b0 + i21 * b1 + i22 * b2);
}

__global__ void fmat_kernel(const float* __restrict__ V, float* __restrict__ F) {
    int t = threadIdx.x;
    if (t >= NPAIRS) return;
    const int pii[NPAIRS] = {0, 0, 0, 1, 1, 2};
    const int pij[NPAIRS] = {1, 2, 3, 2, 3, 3};
    int i = pii[t], j = pij[t];
    float C1[3], C2[3];
    cam_center(V + 16 * i, C1);
    cam_center(V + 16 * j, C2);
    float t0 = C2[0] - C1[0], t1 = C2[1] - C1[1], t2 = C2[2] - C1[2];
    const float* P1 = V + 16 * i;
    const float* P2 = V + 16 * j;
    float R[3][3];
    for (int r = 0; r < 3; r++)
        for (int c = 0; c < 3; c++)
            R[r][c] = P2[4 * r + 0] * P1[4 * c + 0] + P2[4 * r + 1] * P1[4 * c + 1] +
                      P2[4 * r + 2] * P1[4 * c + 2];
    float* Fo = F + 9 * t;
    for (int c = 0; c < 3; c++) {
        Fo[0 * 3 + c] = -t2 * R[1][c] + t1 * R[2][c];
        Fo[1 * 3 + c] =  t2 * R[0][c] - t0 * R[2][c];
        Fo[2 * 3 + c] = -t1 * R[0][c] + t0 * R[1][c];
    }
}

// ---------------------------------------------------------------------------
// Per-point prep: normals, |p|^2, depth loss partials, epipolar partials.
// ---------------------------------------------------------------------------
__global__ __launch_bounds__(256) void prep_kernel(
    const float* __restrict__ pos, const float* __restrict__ rot,
    const float* __restrict__ opac, const float* __restrict__ V,
    const float* __restrict__ F, float* __restrict__ sq, float* __restrict__ nrm,
    float* __restrict__ dpart, float* __restrict__ epart, int N) {
    int n = blockIdx.x * 256 + threadIdx.x;
    float dsum = 0.0f, esum = 0.0f;
    if (n < N) {
        float px = pos[3 * n], py = pos[3 * n + 1], pz = pos[3 * n + 2];
        float op = opac[n];
        float qw = rot[4 * n], qx = rot[4 * n + 1], qy = rot[4 * n + 2], qz = rot[4 * n + 3];
        float inv = rsqrtf(qw * qw + qx * qx + qy * qy + qz * qz);
        qw *= inv; qx *= inv; qy *= inv; qz *= inv;
        nrm[3 * n]     = 2.0f * (qx * qz + qw * qy);
        nrm[3 * n + 1] = 2.0f * (qy * qz - qw * qx);
        nrm[3 * n + 2] = 1.0f - 2.0f * (qx * qx + qy * qy);
        sq[n] = px * px + py * py + pz * pz;
        float cx[NB], cy[NB], cz[NB];
        for (int b = 0; b < NB; b++) {
            const float* M = V + 16 * b;
            cx[b] = M[0] * px + M[1] * py + M[2] * pz + M[3];
            cy[b] = M[4] * px + M[5] * py + M[6] * pz + M[7];
            cz[b] = M[8] * px + M[9] * py + M[10] * pz + M[11];
        }
        for (int b = 0; b < NB - 1; b++) dsum += fabsf(cz[b] * op - cz[b + 1] * op);
        float ptx[NB], pty[NB];
        for (int b = 0; b < NB; b++) {
            float d = fmaxf(cz[b], 1e-8f);
            ptx[b] = cx[b] / d;
            pty[b] = cy[b] / d;
        }
        int pi = 0;
        for (int i = 0; i < NB; i++)
            for (int j = i + 1; j < NB; j++, pi++) {
                const float* Fm = F + 9 * pi;
                float x1 = ptx[i], y1 = pty[i], x2 = ptx[j], y2 = pty[j];
                float l0 = Fm[0] * x1 + Fm[1] * y1 + Fm[2];
                float l1 = Fm[3] * x1 + Fm[4] * y1 + Fm[5];
                float l2 = Fm[6] * x1 + Fm[7] * y1 + Fm[8];
                float err = fabsf(x2 * l0 + y2 * l1 + l2) /
                            (sqrtf(l0 * l0 + l1 * l1) + 1e-8f);
                esum += err;
            }
        esum *= op;
    }
    __shared__ float sred[256];
    int tid = threadIdx.x;
    sred[tid] = dsum;
    __syncthreads();
    for (int s = 128; s > 0; s >>= 1) {
        if (tid < s) sred[tid] += sred[tid + s];
        __syncthreads();
    }
    if (tid == 0) dpart[blockIdx.x] = sred[0];
    __syncthreads();
    sred[tid] = esum;
    __syncthreads();
    for (int s = 128; s > 0; s >>= 1) {
        if (tid < s) sred[tid] += sred[tid + s];
        __syncthreads();
    }
    if (tid == 0) epart[blockIdx.x] = sred[0];
}

// ---------------------------------------------------------------------------
// KNN segment kernel. Grid = (N/16 row tiles) x S column splits, one wave per
// block. WMMA f32 16x16x4 computes 16x16 Gram tiles (K padded 3->4 with 0).
// Column loop is unrolled by 2 with double-buffered, fully branchless B-tile
// loads (lane-indexed component select, one v_cndmask) so global load latency
// hides behind the previous pair's top-K scan without exec-mask juggling.
// Each lane keeps a sorted top-10 over its 8-column half-row; halves merge
// through LDS and lanes 0-15 emit one sorted 10-entry segment per row.
// ---------------------------------------------------------------------------
__global__ __launch_bounds__(32) void knn_kernel(
    const float* __restrict__ pos, const float* __restrict__ sq,
    float* __restrict__ knnd, int* __restrict__ knni, int N, int S,
    int tilesPerSplit) {
    __shared__ float lds_dot[2 * 16 * 17];
    __shared__ float lds_sqc[32];
    __shared__ float lds_bd[32 * KNN];
    __shared__ int lds_bi[32 * KNN];

    const int lane = threadIdx.x;
    const int nl = lane & 15;
    const int half = lane >> 4;
    const int rt = blockIdx.x / S;
    const int cs = blockIdx.x - rt * S;
    const int rb = rt * 16;
    const int i = rb + nl;  // query point for this lane's row duty
    const float sqr = sq[i];

    // A-matrix (16x4 f32): lanes 0-15 hold (K0,K1)=(x,y) of row M=lane;
    // lanes 16-31 hold (K2,K3)=(z,0) of row M=lane-16.  (ISA 7.12.2)
    // Branchless: component 2*half selects x vs z; cndmask zeroes K3.
    v2f amat;
    {
        float a0 = pos[3 * i + 2 * half];
        float a1 = pos[3 * i + 1];
        amat.x = a0;
        amat.y = half ? 0.0f : a1;
    }

    float bd[KNN];
    int bi[KNN];
#pragma unroll
    for (int k = 0; k < KNN; k++) { bd[k] = 3.0e38f; bi[k] = -1; }

    // Branchless load of a B-tile pair + per-tile sq cache value.
    // b0/b1 need columns cb..cb+15 / cb+16..cb+31 in BOTH halves;
    // sq cache: half-0 lanes fetch tile0's column sq, half-1 tile1's.
#define LOADB2(cb, bm0, bm1, sc)                                                \
    do {                                                                        \
        int c0_ = (cb) + nl;                                                    \
        int c1_ = c0_ + 16;                                                     \
        float x0_ = pos[3 * c0_ + 2 * half];                                    \
        float y0_ = pos[3 * c0_ + 1];                                           \
        float x1_ = pos[3 * c1_ + 2 * half];                                    \
        float y1_ = pos[3 * c1_ + 1];                                           \
        sc = sq[(cb) + 16 * half + nl];                                         \
        bm0.x = x0_; bm0.y = half ? 0.0f : y0_;                                 \
        bm1.x = x1_; bm1.y = half ? 0.0f : y1_;                                 \
    } while (0)

#define SCAN_TILE(tsel, cbase)                                                  \
    do {                                                                        \
        _Pragma("unroll") for (int u = 0; u < 8; ++u) {                         \
            int nn = half * 8 + u;                                              \
            float d2 = sqr + lds_sqc[(tsel)*16 + nn] -                          \
                       2.0f * lds_dot[(tsel)*272 + nl * 17 + nn];               \
            int j = (cbase) + nn;                                               \
            if (j != i && d2 < bd[KNN - 1]) {                                   \
                bd[KNN - 1] = d2;                                               \
                bi[KNN - 1] = j;                                                \
                _Pragma("unroll") for (int q = KNN - 1; q > 0; --q) {           \
                    if (bd[q] < bd[q - 1]) {                                    \
                        float td = bd[q]; bd[q] = bd[q - 1]; bd[q - 1] = td;    \
                        int ti = bi[q]; bi[q] = bi[q - 1]; bi[q - 1] = ti;      \
                    }                                                           \
                }                                                               \
            }                                                                   \
        }                                                                       \
    } while (0)

    const int ct0 = cs * tilesPerSplit;
    const int ct1 = ct0 + tilesPerSplit;
    int ct = ct0;

    v2f b0, b1;
    float scv = 0.0f;
    if (ct + 1 < ct1) LOADB2(ct * 16, b0, b1, scv);

    // Main pairs (unroll-by-2, double buffered).
    for (; ct + 1 < ct1; ct += 2) {
        const int cbase = ct * 16;
        v8f c0 = {}, c1 = {};
        c0 = __builtin_amdgcn_wmma_f32_16x16x4_f32(false, amat, false, b0,
                                                   (short)0, c0, false, false);
        c1 = __builtin_amdgcn_wmma_f32_16x16x4_f32(false, amat, false, b1,
                                                   (short)0, c1, false, false);
        lds_sqc[16 * half + nl] = scv;  // all 32 lanes: tile0 by half0, tile1 by half1
        // C/D layout: VGPR v, lanes 0-15 -> (M=v, N=lane); lanes 16-31 -> M=v+8.
#pragma unroll
        for (int v = 0; v < 8; ++v) {
            lds_dot[(v + 8 * half) * 17 + nl] = c0[v];
            lds_dot[272 + (v + 8 * half) * 17 + nl] = c1[v];
        }
        // Prefetch next pair's operands; latency hides behind the scan below.
        if (ct + 3 < ct1) LOADB2(cbase + 32, b0, b1, scv);
        __syncthreads();
        SCAN_TILE(0, cbase);
        SCAN_TILE(1, cbase + 16);
        __syncthreads();
    }
    // Odd tail tile (only if tilesPerSplit is odd).
    if (ct < ct1) {
        const int cbase = ct * 16;
        int c0_ = cbase + nl;
        v2f bt;
        {
            float x0_ = pos[3 * c0_ + 2 * half];
            float y0_ = pos[3 * c0_ + 1];
            bt.x = x0_;
            bt.y = half ? 0.0f : y0_;
        }
        float stv = sq[c0_];
        v8f c0 = {};
        c0 = __builtin_amdgcn_wmma_f32_16x16x4_f32(false, amat, false, bt,
                                                   (short)0, c0, false, false);
        if (half == 0) lds_sqc[nl] = stv;
#pragma unroll
        for (int v = 0; v < 8; ++v) lds_dot[(v + 8 * half) * 17 + nl] = c0[v];
        __syncthreads();
        SCAN_TILE(0, cbase);
        __syncthreads();
    }

    // Merge the two sorted half-lists of each row, emit segment to workspace.
#pragma unroll
    for (int k = 0; k < KNN; k++) {
        lds_bd[lane * KNN + k] = bd[k];
        lds_bi[lane * KNN + k] = bi[k];
    }
    __syncthreads();
    if (lane < 16) {
        int a = 0, b = 0;
        float* od = knnd + ((size_t)i * S + cs) * KNN;
        int* oi = knni + ((size_t)i * S + cs) * KNN;
#pragma unroll
        for (int k = 0; k < KNN; k++) {
            float da = lds_bd[lane * KNN + a], db = lds_bd[(lane + 16) * KNN + b];
            int ja = lds_bi[lane * KNN + a], jb = lds_bi[(lane + 16) * KNN + b];
            bool takeA = (da < db) || (da == db && ja < jb);
            float md = takeA ? da : db;
            int mj = takeA ? ja : jb;
            if (takeA) a++; else b++;
            od[k] = md;
            oi[k] = mj;
        }
    }
#undef LOADB2
#undef SCAN_TILE
}

// ---------------------------------------------------------------------------
// S-way merge of sorted segments + normal-cosine loss (one thread per row).
// ---------------------------------------------------------------------------
__global__ __launch_bounds__(256) void merge_kernel(
    const float* __restrict__ knnd, const int* __restrict__ knni,
    const float* __restrict__ nrm, const float* __restrict__ opac,
    float* __restrict__ npart, int N, int S) {
    int i = blockIdx.x * 256 + threadIdx.x;
    float val = 0.0f;
    if (i < N) {
        int ptr[SMAX];
        for (int s = 0; s < SMAX; s++) ptr[s] = 0;
        const float* dbase = knnd + (size_t)i * S * KNN;
        const int* ibase = knni + (size_t)i * S * KNN;
        float nx = nrm[3 * i], ny = nrm[3 * i + 1], nz = nrm[3 * i + 2];
        float acc = 0.0f;
        for (int k = 0; k < KNN; k++) {
            float best = 3.4e38f;
            int bj = 0x7fffffff, bs = 0;
            for (int s = 0; s < S; s++) {
                if (ptr[s] < KNN) {
                    float d = dbase[s * KNN + ptr[s]];
                    int j = ibase[s * KNN + ptr[s]];
                    if (d < best || (d == best && j < bj)) { best = d; bj = j; bs = s; }
                }
            }
            ptr[bs]++;
            if (bj >= 0 && bj < N)
                acc += 1.0f - (nx * nrm[3 * bj] + ny * nrm[3 * bj + 1] + nz * nrm[3 * bj + 2]);
        }
        val = opac[i] * (acc / (float)KNN);
    }
    __shared__ float sred[256];
    int tid = threadIdx.x;
    sred[tid] = val;
    __syncthreads();
    for (int s = 128; s > 0; s >>= 1) {
        if (tid < s) sred[tid] += sred[tid + s];
        __syncthreads();
    }
    if (tid == 0) npart[blockIdx.x] = sred[0];
}

// ---------------------------------------------------------------------------
// Deterministic final reduction + normalization.
// ---------------------------------------------------------------------------
__global__ void final_kernel(const float* __restrict__ dpart,
                             const float* __restrict__ epart,
                             const float* __restrict__ npart, float* __restrict__ out,
                             int nblk_prep, int nblk_merge, int N) {
    if (threadIdx.x != 0 || blockIdx.x != 0) return;
    float ds = 0.0f, es = 0.0f, ns = 0.0f;
    for (int k = 0; k < nblk_prep; k++) { ds += dpart[k]; es += epart[k]; }
    for (int k = 0; k < nblk_merge; k++) ns += npart[k];
    float depth_loss = ds / ((float)N * (float)(NB - 1));
    float epi_loss = es / (float)NPAIRS;
    float normal_loss = ns / (float)N;
    out[0] = depth_loss + normal_loss + epi_loss;
}

extern "C" void kernel_launch(void* const* d_in, const int* in_sizes, int n_in,
                              void* d_out, int out_size, void* d_ws, size_t ws_size,
                              hipStream_t stream) {
    const float* pos = (const float*)d_in[0];   // N x 3
    const float* rot = (const float*)d_in[1];   // N x 4
    const float* opac = (const float*)d_in[2];  // N x 1
    const float* V = (const float*)d_in[3];     // B x 4 x 4
    float* out = (float*)d_out;
    float* ws = (float*)d_ws;
    const int N = in_sizes[0] / 3;

    const int nblk_prep = (N + 255) / 256;
    const int nblk_merge = (N + 255) / 256;
    const int ntiles = N / 16;

    // Column-split factor: more blocks -> better SIMD fill. Constrained by
    // workspace (segments cost 2*KNN*S words per row) and divisibility.
    size_t base_words = 64 + (size_t)4 * N + (size_t)2 * nblk_prep + nblk_merge;
    int S = 4;
    while (S > 1 &&
           ((base_words + (size_t)N * 2 * KNN * S) * 4 > ws_size || (ntiles % S) != 0))
        S >>= 1;
    const int tilesPerSplit = ntiles / S;

    // Workspace layout (4-byte words).
    size_t off = 0;
    float* F = ws + off;     off += 64;
    float* sq = ws + off;    off += (size_t)N;
    float* nrm = ws + off;   off += (size_t)3 * N;
    float* dpart = ws + off; off += (size_t)nblk_prep;
    float* epart = ws + off; off += (size_t)nblk_prep;
    float* npart = ws + off; off += (size_t)nblk_merge;
    float* knnd = ws + off;  off += (size_t)N * KNN * S;
    int* knni = (int*)(ws + off);

    hipLaunchKernelGGL(fmat_kernel, dim3(1), dim3(32), 0, stream, V, F);
    hipLaunchKernelGGL(prep_kernel, dim3(nblk_prep), dim3(256), 0, stream,
                       pos, rot, opac, V, F, sq, nrm, dpart, epart, N);
    hipLaunchKernelGGL(knn_kernel, dim3((N / 16) * S), dim3(32), 0, stream,
                       pos, sq, knnd, knni, N, S, tilesPerSplit);
    hipLaunchKernelGGL(merge_kernel, dim3(nblk_merge), dim3(256), 0, stream,
                       knnd, knni, nrm, opac, npart, N, S);
    hipLaunchKernelGGL(final_kernel, dim3(1), dim3(1), 0, stream,
                       dpart, epart, npart, out, nblk_prep, nblk_merge, N);
}